// MDTA_747324310132
// MI455X (gfx1250) — compile-verified
//
#include <hip/hip_runtime.h>
#include <hip/hip_bf16.h>
#include <stdint.h>

typedef float v2f __attribute__((ext_vector_type(2)));
typedef float v8f __attribute__((ext_vector_type(8)));

#define GK 192          // inner channel dim for both big GEMMs
#define GN 65536        // pixels per image
#define BM 64
#define BN 64
#define WSL 194         // W-slab LDS row stride (even -> 8B-aligned v2f A reads, spread banks)
#define XS  68          // X-tile LDS row stride (mult of 4 -> 16B-aligned async B128 stores)
#define QS  68          // gram tile LDS row stride

// ---- CDNA5 async global->LDS path (ASYNCcnt) -------------------------------
__device__ __forceinline__ unsigned lds_off(const void* p) {
    // generic shared pointer: low 32 bits are the LDS offset (aperture in high half)
    return (unsigned)(uintptr_t)p;
}
__device__ __forceinline__ void async_ld_b128(unsigned loff, const void* gaddr) {
    asm volatile("global_load_async_to_lds_b128 %0, %1, off"
                 :: "v"(loff), "v"(gaddr) : "memory");
}
__device__ __forceinline__ void wait_async_0() { asm volatile("s_wait_asynccnt 0x0" ::: "memory"); }
__device__ __forceinline__ void wait_async_1() { asm volatile("s_wait_asynccnt 0x1" ::: "memory"); }
__device__ __forceinline__ void wait_async_6() { asm volatile("s_wait_asynccnt 0x6" ::: "memory"); }

// ---------------------------------------------------------------------------
// Generic GEMM: Y[b] = W[b] (M x 192) @ X[b] (192 x 65536), fp32 WMMA 16x16x4.
// W slab (64 x 192) resident in LDS; X double-buffered via async-to-LDS B128.
// ---------------------------------------------------------------------------
__global__ __launch_bounds__(256)
void gemm192_kernel(const float* __restrict__ W, long wstride,
                    const float* __restrict__ X, long xstride,
                    float* __restrict__ Y, long ystride)
{
    __shared__ float wslab[BM * WSL];     // [64][194]
    __shared__ float xt[2][16 * XS];      // double-buffered [16 k][68] tiles

    const int tid  = threadIdx.x;
    const int lane = tid & 31;
    const int wave = tid >> 5;
    const int half = lane >> 4;
    const int l16  = lane & 15;
    const int wm   = wave >> 1;   // M subtile 0..3
    const int wn   = wave & 1;    // N half 0..1 (32 cols each)

    const int b  = blockIdx.z;
    const int m0 = blockIdx.y * BM;
    const int n0 = blockIdx.x * BN;

    const float* Wb = W + (long)b * wstride;
    const float* Xb = X + (long)b * xstride;
    float*       Yb = Y + (long)b * ystride;

    // X staging: 16x64 floats = 256 float4 -> exactly 1 async B128 per thread
    const int skr = tid >> 4;             // k row 0..15
    const int sc4 = (tid & 15) * 4;       // n col (float4 granularity)

    // Prologue: issue tile k0=0 into buffer 0
    async_ld_b128(lds_off(&xt[0][skr * XS + sc4]),
                  Xb + (long)skr * GN + n0 + sc4);

    // Load W slab [64][192] once
    #pragma unroll
    for (int i = 0; i < 12; ++i) {
        int id = tid + i * 256;           // 0..3071 float4s
        int r  = id / 48;
        int c4 = (id % 48) * 4;
        float4 v = *(const float4*)(Wb + (long)(m0 + r) * GK + c4);
        wslab[r * WSL + c4 + 0] = v.x;
        wslab[r * WSL + c4 + 1] = v.y;
        wslab[r * WSL + c4 + 2] = v.z;
        wslab[r * WSL + c4 + 3] = v.w;
    }

    v8f acc[2] = {};
    int buf = 0;
    for (int k0 = 0; k0 < GK; k0 += 16) {
        if (k0 + 16 < GK) {               // prefetch next tile into other buffer
            async_ld_b128(lds_off(&xt[buf ^ 1][skr * XS + sc4]),
                          Xb + (long)(k0 + 16 + skr) * GN + n0 + sc4);
            wait_async_1();               // in-order: current tile complete
        } else {
            wait_async_0();
        }
        __syncthreads();                  // all waves' async copies + wslab visible

        #pragma unroll
        for (int ks = 0; ks < 4; ++ks) {
            int kl = ks * 4 + 2 * half;   // A/B vgpr j -> K = kl + j
            v2f a = *(const v2f*)&wslab[(wm * 16 + l16) * WSL + k0 + kl];
            #pragma unroll
            for (int nt = 0; nt < 2; ++nt) {
                int n = wn * 32 + nt * 16 + l16;
                v2f bf = { xt[buf][(kl + 0) * XS + n],
                           xt[buf][(kl + 1) * XS + n] };
                acc[nt] = __builtin_amdgcn_wmma_f32_16x16x4_f32(
                    false, a, false, bf, (short)0, acc[nt], false, false);
            }
        }
        __syncthreads();                  // done reading buf before it is refilled
        buf ^= 1;
    }

    // Store: C/D vgpr i -> M = i + 8*half, N = l16
    #pragma unroll
    for (int nt = 0; nt < 2; ++nt) {
        int n     = n0 + wn * 32 + nt * 16 + l16;
        int mbase = m0 + wm * 16 + 8 * half;
        #pragma unroll
        for (int i = 0; i < 8; ++i)
            Yb[(long)(mbase + i) * GN + n] = acc[nt][i];
    }
}

// ---------------------------------------------------------------------------
// Per-row inverse L2 norms for q,k rows (o in [0,384) of qkv). Block per row.
// ---------------------------------------------------------------------------
__global__ __launch_bounds__(256)
void rownorm_kernel(const float* __restrict__ qkv, float* __restrict__ invn)
{
    int row = blockIdx.x;                  // b*384 + o
    int b = row / 384, o = row % 384;
    const float4* p4 = (const float4*)(qkv + ((long)b * 576 + o) * GN);
    float s = 0.f;
    for (int i = threadIdx.x; i < GN / 4; i += 256) {
        float4 v = p4[i];
        s += v.x * v.x + v.y * v.y + v.z * v.z + v.w * v.w;
    }
    __shared__ float red[256];
    red[threadIdx.x] = s;
    __syncthreads();
    for (int off = 128; off > 0; off >>= 1) {
        if (threadIdx.x < off) red[threadIdx.x] += red[threadIdx.x + off];
        __syncthreads();
    }
    if (threadIdx.x == 0)
        invn[row] = 1.0f / fmaxf(sqrtf(red[0]), 1e-12f);
}

__global__ void zero_kernel(float* __restrict__ p, int n)
{
    int i = blockIdx.x * 256 + threadIdx.x;
    if (i < n) p[i] = 0.f;
}

// ---------------------------------------------------------------------------
// Gram: S[b,h,c,d] = sum_n q[c,n]*k[d,n]; K=65536 split over gridDim.y chunks
// with f32 atomic accumulation. Tiles double-buffered via async-to-LDS; the 8
// zero-pad rows (24..31) are written once and never touched by async copies.
// ---------------------------------------------------------------------------
#define TKG 64
#define GRAM_ITER 32   // chunk = 64*32 = 2048 K per block

__global__ __launch_bounds__(128)
void gram_kernel(const float* __restrict__ qkv, float* __restrict__ S)
{
    __shared__ float qt[2][32 * QS];
    __shared__ float kt[2][32 * QS];

    int bh = blockIdx.x;                   // b*8 + h
    int b = bh >> 3, h = bh & 7;
    long nbase0 = (long)blockIdx.y * (TKG * GRAM_ITER);

    int tid = threadIdx.x, lane = tid & 31, wave = tid >> 5;
    int half = lane >> 4, l16 = lane & 15;
    int mt = wave >> 1, nt = wave & 1;     // 2x2 tiles of 16 covering 24x24

    const float* qbase = qkv + ((long)b * 576 + h * 24) * GN;
    const float* kbase = qbase + 192L * GN;

    // Zero pad rows 24..31 of all buffers (read-only thereafter)
    for (int i = tid; i < 8 * QS; i += 128) {
        qt[0][24 * QS + i] = 0.f;
        qt[1][24 * QS + i] = 0.f;
        kt[0][24 * QS + i] = 0.f;
        kt[1][24 * QS + i] = 0.f;
    }

    // Prologue: issue tile it=0 into buffer 0 (24 rows x 16 f4 per matrix,
    // 3 B128 issues per thread per matrix -> 6 per wave per tile, uniform)
    #pragma unroll
    for (int i = 0; i < 3; ++i) {
        int id = tid + i * 128;            // 0..383
        int r  = id >> 4;                  // 0..23
        int c4 = (id & 15) * 4;
        async_ld_b128(lds_off(&qt[0][r * QS + c4]), qbase + (long)r * GN + nbase0 + c4);
        async_ld_b128(lds_off(&kt[0][r * QS + c4]), kbase + (long)r * GN + nbase0 + c4);
    }

    v8f acc = {};
    int buf = 0;
    for (int it = 0; it < GRAM_ITER; ++it) {
        if (it + 1 < GRAM_ITER) {
            long nb = nbase0 + (long)(it + 1) * TKG;
            #pragma unroll
            for (int i = 0; i < 3; ++i) {
                int id = tid + i * 128;
                int r  = id >> 4;
                int c4 = (id & 15) * 4;
                async_ld_b128(lds_off(&qt[buf ^ 1][r * QS + c4]), qbase + (long)r * GN + nb + c4);
                async_ld_b128(lds_off(&kt[buf ^ 1][r * QS + c4]), kbase + (long)r * GN + nb + c4);
            }
            wait_async_6();                // current tile's 6 copies complete
        } else {
            wait_async_0();
        }
        __syncthreads();

        #pragma unroll
        for (int kk = 0; kk < TKG / 4; ++kk) {
            int kl = kk * 4 + 2 * half;
            v2f a  = *(const v2f*)&qt[buf][(mt * 16 + l16) * QS + kl];
            v2f bf = *(const v2f*)&kt[buf][(nt * 16 + l16) * QS + kl];
            acc = __builtin_amdgcn_wmma_f32_16x16x4_f32(
                false, a, false, bf, (short)0, acc, false, false);
        }
        __syncthreads();
        buf ^= 1;
    }

    #pragma unroll
    for (int i = 0; i < 8; ++i) {
        int c = mt * 16 + 8 * half + i;
        int d = nt * 16 + l16;
        if (c < 24 && d < 24)
            atomicAdd(&S[((long)bh * 24 + c) * 24 + d], acc[i]);
    }
}

// ---------------------------------------------------------------------------
// Softmax over d with l2-norm scaling and temperature. One wave per (b,h,c).
// ---------------------------------------------------------------------------
__global__ __launch_bounds__(32)
void softmax_kernel(const float* __restrict__ S, const float* __restrict__ invn,
                    const float* __restrict__ temp, float* __restrict__ attn)
{
    int idx = blockIdx.x;                  // (b*8 + h)*24 + c
    int c = idx % 24;
    int bh = idx / 24;
    int h = bh & 7, b = bh >> 3;
    int d = threadIdx.x;
    bool valid = d < 24;

    float iq = invn[b * 384 + h * 24 + c];
    float t  = temp[h];
    float v  = -1e30f;
    if (valid) {
        float ik = invn[b * 384 + 192 + h * 24 + d];
        v = S[((long)bh * 24 + c) * 24 + d] * iq * ik * t;
    }
    float m = v;
    for (int off = 16; off > 0; off >>= 1) m = fmaxf(m, __shfl_xor(m, off, 32));
    float e = valid ? __expf(v - m) : 0.f;
    float s = e;
    for (int off = 16; off > 0; off >>= 1) s += __shfl_xor(s, off, 32);
    if (valid) attn[((long)bh * 24 + c) * 24 + d] = e / s;
}

// ---------------------------------------------------------------------------
// M_b = W_out @ blockdiag(attn_b): tiny per-batch 192x192 with K=24 per entry.
// ---------------------------------------------------------------------------
__global__ __launch_bounds__(256)
void mix_kernel(const float* __restrict__ wout, const float* __restrict__ attn,
                float* __restrict__ Mb)
{
    int b = blockIdx.y;
    int idx = blockIdx.x * 256 + threadIdx.x;   // 0..36863
    int o = idx / 192, g = idx % 192;
    int h = g / 24, d = g % 24;
    const float* wrow = wout + o * 192 + h * 24;
    const float* acol = attn + ((long)(b * 8 + h) * 24) * 24 + d;
    float s = 0.f;
    #pragma unroll
    for (int cc = 0; cc < 24; ++cc) s += wrow[cc] * acol[cc * 24];
    Mb[((long)b * 192 + o) * 192 + g] = s;
}

// ---------------------------------------------------------------------------
extern "C" void kernel_launch(void* const* d_in, const int* in_sizes, int n_in,
                              void* d_out, int out_size, void* d_ws, size_t ws_size,
                              hipStream_t stream)
{
    const float* x     = (const float*)d_in[0];   // (4,192,256,256)
    const float* w_qkv = (const float*)d_in[1];   // (576,192)
    const float* w_out = (const float*)d_in[2];   // (192,192)
    const float* temp  = (const float*)d_in[3];   // (8,1,1)
    float* out = (float*)d_out;                   // (4,192,256,256)

    float* ws   = (float*)d_ws;
    float* qkv  = ws;                              // 4*576*65536 floats (604 MB)
    float* invn = qkv + 4L * 576 * GN;             // 1536 floats (q,k row inv-norms)
    float* S    = invn + 4 * 384;                  // 4*8*24*24 Gram
    float* attn = S + 4 * 8 * 24 * 24;             // softmaxed attention
    float* Mb   = attn + 4 * 8 * 24 * 24;          // 4*192*192 fused W_out@A_b

    // 1. qkv = W_qkv @ x   (M=576, shared weights -> wstride 0)
    gemm192_kernel<<<dim3(GN / BN, 576 / BM, 4), 256, 0, stream>>>(
        w_qkv, 0L, x, 192L * GN, qkv, 576L * GN);

    // 2. inverse L2 norms of q,k rows
    rownorm_kernel<<<4 * 384, 256, 0, stream>>>(qkv, invn);

    // 3. zero Gram accumulator (fresh every call: atomics + graph replay)
    zero_kernel<<<(4 * 8 * 24 * 24 + 255) / 256, 256, 0, stream>>>(S, 4 * 8 * 24 * 24);

    // 4. Gram S = q @ k^T per (b,h), K split over 32 chunks
    gram_kernel<<<dim3(32, GN / (TKG * GRAM_ITER)), 128, 0, stream>>>(qkv, S);

    // 5. softmax(norm-scaled, temperature)
    softmax_kernel<<<4 * 8 * 24, 32, 0, stream>>>(S, invn, temp, attn);

    // 6. M_b = W_out @ blockdiag(attn)
    mix_kernel<<<dim3(144, 4), 256, 0, stream>>>(w_out, attn, Mb);

    // 7. out = M_b @ v   (v = qkv rows 384..575 per batch)
    gemm192_kernel<<<dim3(GN / BN, 192 / BM, 4), 256, 0, stream>>>(
        Mb, 192L * 192, qkv + 384L * GN, 576L * GN, out, 192L * GN);
}